// LocallyDense_62311385530546
// MI455X (gfx1250) — compile-verified
//
#include <hip/hip_runtime.h>

typedef __attribute__((ext_vector_type(2))) float v2f;
typedef __attribute__((ext_vector_type(8))) float v8f;

#define NEG_SLOPE 0.2f

// Problem dimensions (fixed by the reference)
#define B_  32
#define N_  1024
#define F_  512
#define G_  16
#define S_  64
#define D_  256
#define KW  32     // conv window
#define CO  512    // conv output channels
#define NO  993    // N - KW + 1 (VALID conv output length)

// ---------------------------------------------------------------------------
// Stage 1: mid[b, g*S+s, d] = sum_f x[b, idx[g,s], f] * W[g, f, d] + bias[g,d]
// One wave32 computes a full 64x16 strip (4 row tiles) of a (b,g) block,
// reusing each B panel load across 4 WMMAs. Full fp32 via
// V_WMMA_F32_16X16X4_F32 (problem is HBM-bound; no need to drop precision).
// ---------------------------------------------------------------------------
__global__ __launch_bounds__(256) void stage1_grouped_gemm(
    const float* __restrict__ x, const int* __restrict__ idx,
    const float* __restrict__ W, const float* __restrict__ bias,
    float* __restrict__ mid)
{
    const int wave = blockIdx.x * (blockDim.x >> 5) + (threadIdx.x >> 5);
    const int lane = threadIdx.x & 31;
    const int half = lane >> 4;   // K half: {0,1} vs {2,3}
    const int lo   = lane & 15;   // row (A) / col (B) within tile

    // 8192 waves: 16 col tiles (D) x 16 groups x 32 batches
    const int colTile = wave & 15;
    const int g       = (wave >> 4) & 15;
    const int b       = wave >> 8;

    // A: gathered row pointers, one per 16-row sub-tile (s = t*16 + lo)
    const float* __restrict__ xrow[4];
#pragma unroll
    for (int t = 0; t < 4; ++t) {
        const int row = idx[g * S_ + t * 16 + lo];
        xrow[t] = x + ((size_t)b * N_ + row) * F_;
    }

    // B: coalesced column within W[g]
    const int col = colTile * 16 + lo;
    const float* __restrict__ wg = W + (size_t)g * F_ * D_ + col;

    v8f c0 = {}, c1 = {}, c2 = {}, c3 = {};
#pragma unroll 4
    for (int k = 0; k < F_; k += 4) {
        const int ka = k + 2 * half;
        v2f bm;                                     // B[ka..ka+1][col]
        bm.x = wg[(size_t)(ka + 0) * D_];
        bm.y = wg[(size_t)(ka + 1) * D_];
        v2f a0 = *(const v2f*)(xrow[0] + ka);
        v2f a1 = *(const v2f*)(xrow[1] + ka);
        v2f a2 = *(const v2f*)(xrow[2] + ka);
        v2f a3 = *(const v2f*)(xrow[3] + ka);
        c0 = __builtin_amdgcn_wmma_f32_16x16x4_f32(false, a0, false, bm, (short)0, c0, false, false);
        c1 = __builtin_amdgcn_wmma_f32_16x16x4_f32(false, a1, false, bm, (short)0, c1, false, false);
        c2 = __builtin_amdgcn_wmma_f32_16x16x4_f32(false, a2, false, bm, (short)0, c2, false, false);
        c3 = __builtin_amdgcn_wmma_f32_16x16x4_f32(false, a3, false, bm, (short)0, c3, false, false);
    }

    const float bb = bias[g * D_ + col];
    float* out = mid + ((size_t)b * (G_ * S_) + (size_t)g * S_) * D_ + col;
    const v8f acc[4] = {c0, c1, c2, c3};
#pragma unroll
    for (int t = 0; t < 4; ++t) {
#pragma unroll
        for (int r = 0; r < 8; ++r) {
            const int m = t * 16 + r + 8 * half;    // C/D row layout
            out[(size_t)m * D_] = acc[t][r] + bb;
        }
    }
}

// ---------------------------------------------------------------------------
// Stage 2: implicit-GEMM 1D conv + bias + leaky ReLU.
// y[b, n, o] = act( sum_{w,d} mid[b, n+w, d] * conv_w[w, d, o] + conv_b[o] )
// One wave32 per 64x16 (n, o) strip; K = 32*256 = 8192, stepped 4 at a time.
// conv_w panel loads are reused across 4 row tiles. Row clamp keeps EXEC
// all-ones for WMMA; out-of-range rows masked only at the store.
// ---------------------------------------------------------------------------
__global__ __launch_bounds__(256) void stage2_conv_gemm(
    const float* __restrict__ mid, const float* __restrict__ cw,
    const float* __restrict__ cb, float* __restrict__ y)
{
    const int wave = blockIdx.x * (blockDim.x >> 5) + (threadIdx.x >> 5);
    const int lane = threadIdx.x & 31;
    const int half = lane >> 4;
    const int lo   = lane & 15;

    // 16384 waves: 32 col tiles (CO) x 16 row groups (4 tiles each) x 32 batches
    const int colTile = wave & 31;
    const int rg      = (wave >> 5) & 15;
    const int b       = wave >> 9;

    const int nGroupBase = rg * 64;
    const float* __restrict__ arow[4];
#pragma unroll
    for (int t = 0; t < 4; ++t) {
        int n = nGroupBase + t * 16 + lo;
        if (n > NO - 1) n = NO - 1;                 // clamp: loads in-bounds,
                                                    // EXEC all-ones for WMMA
        arow[t] = mid + ((size_t)b * N_ + n) * D_;
    }
    const int o = colTile * 16 + lo;

    v8f c0 = {}, c1 = {}, c2 = {}, c3 = {};
    for (int w = 0; w < KW; ++w) {
        const size_t woff = (size_t)w * D_;
        const float* __restrict__ bw = cw + woff * CO + o;
        if (w + 1 < KW)                             // gfx1250 global_prefetch_b8
            __builtin_prefetch(bw + (size_t)D_ * CO, 0, 1);
#pragma unroll 2
        for (int d = 0; d < D_; d += 4) {
            const int dd = d + 2 * half;
            v2f bm;                                 // conv_w[w, dd..dd+1, o]
            bm.x = bw[(size_t)(dd + 0) * CO];
            bm.y = bw[(size_t)(dd + 1) * CO];
            v2f a0 = *(const v2f*)(arow[0] + woff + dd);
            v2f a1 = *(const v2f*)(arow[1] + woff + dd);
            v2f a2 = *(const v2f*)(arow[2] + woff + dd);
            v2f a3 = *(const v2f*)(arow[3] + woff + dd);
            c0 = __builtin_amdgcn_wmma_f32_16x16x4_f32(false, a0, false, bm, (short)0, c0, false, false);
            c1 = __builtin_amdgcn_wmma_f32_16x16x4_f32(false, a1, false, bm, (short)0, c1, false, false);
            c2 = __builtin_amdgcn_wmma_f32_16x16x4_f32(false, a2, false, bm, (short)0, c2, false, false);
            c3 = __builtin_amdgcn_wmma_f32_16x16x4_f32(false, a3, false, bm, (short)0, c3, false, false);
        }
    }

    const float bb = cb[o];
    const v8f acc[4] = {c0, c1, c2, c3};
#pragma unroll
    for (int t = 0; t < 4; ++t) {
#pragma unroll
        for (int r = 0; r < 8; ++r) {
            const int nn = nGroupBase + t * 16 + r + 8 * half;
            if (nn < NO) {
                float v = acc[t][r] + bb;
                y[((size_t)b * NO + nn) * CO + o] = (v >= 0.0f) ? v : NEG_SLOPE * v;
            }
        }
    }
}

// ---------------------------------------------------------------------------
extern "C" void kernel_launch(void* const* d_in, const int* in_sizes, int n_in,
                              void* d_out, int out_size, void* d_ws, size_t ws_size,
                              hipStream_t stream) {
    (void)in_sizes; (void)n_in; (void)out_size; (void)ws_size;
    const float* x   = (const float*)d_in[0];
    const int*   idx = (const int*)d_in[1];
    const float* W   = (const float*)d_in[2];
    const float* bia = (const float*)d_in[3];
    const float* cw  = (const float*)d_in[4];
    const float* cb  = (const float*)d_in[5];
    float*       y   = (float*)d_out;
    float*       mid = (float*)d_ws;   // 32*1024*256 fp32 = 32 MB scratch

    // Stage 1: 8192 waves / 8 waves per 256-thread block = 1024 blocks
    stage1_grouped_gemm<<<1024, 256, 0, stream>>>(x, idx, W, bia, mid);
    // Stage 2: 16384 waves / 8 = 2048 blocks
    stage2_conv_gemm<<<2048, 256, 0, stream>>>(mid, cw, cb, y);
}